// FPLUT_1185410973916
// MI455X (gfx1250) — compile-verified
//
#include <hip/hip_runtime.h>

typedef _Float16 half_t;
typedef __attribute__((ext_vector_type(4))) float f32x4;
typedef __attribute__((ext_vector_type(4))) unsigned int u32x4;
typedef __attribute__((ext_vector_type(8))) int i32x8;
typedef __attribute__((ext_vector_type(4))) int i32x4;

#define NUM_POINTS 4
#define TABLE_N 35
#define NUM_TABLES 10
#define TAB_ENTRIES 65536
#define TAB_BYTES (TAB_ENTRIES * 2)   // 128 KB fp16 table (fits CDNA5 320KB LDS)

// ---------------------------------------------------------------------------
// Exact reference piecewise evaluation for one fp16 value (used by pre-kernel
// and by the fallback kernel).
// ---------------------------------------------------------------------------
__device__ __forceinline__ float gelu_piecewise(
    float xf, const float* __restrict__ cutf,
    const float* __restrict__ tbl, const unsigned* __restrict__ cs_pk,
    half_t t0h, half_t tlh)
{
  float xcl = __builtin_fminf(__builtin_fmaxf(xf, cutf[0]), cutf[10]);
  half_t xc = (half_t)xcl;
  xc = (xf != xf) ? (half_t)0.0f : xc;
  float x32 = (float)xc;

  int ci = 0;
#pragma unroll
  for (int j = 1; j <= 9; ++j) ci += (x32 >= cutf[j]) ? 1 : 0;

  unsigned cs = cs_pk[ci];
  half_t cuth = __builtin_bit_cast(half_t, (unsigned short)(cs & 0xFFFFu));
  half_t sclh = __builtin_bit_cast(half_t, (unsigned short)(cs >> 16));

  half_t temp = (half_t)((xc - cuth) * sclh);
  int idx = (int)(float)temp;                 // temp >= 0 -> trunc == floor
  idx = ((ci == NUM_TABLES - 1) & (idx == 1)) ? 0 : idx;
  half_t dec = temp - (half_t)idx;
  int ti = (ci == 0) ? idx : (1 + (ci - 1) * NUM_POINTS + idx);

  float left  = tbl[ti];
  float right = tbl[ti + 1];
  half_t y = (half_t)left + ((half_t)right - (half_t)left) * dec;

  y = (x32 <= cutf[0])  ? t0h : y;
  y = (x32 >= cutf[10]) ? tlh : y;
  return (float)y;
}

// ---------------------------------------------------------------------------
// Pre-kernel: build direct fp16->fp16 table (one entry per fp16 bit pattern).
// 65536 threads total; cost is negligible vs the 67M-element stream.
// ---------------------------------------------------------------------------
__global__ __launch_bounds__(256) void build_tab_kernel(
    const half_t* __restrict__ cutp,
    const float* __restrict__ tablep,
    const float* __restrict__ scalep,
    unsigned short* __restrict__ tab)
{
  int b = blockIdx.x * blockDim.x + threadIdx.x;
  if (b >= TAB_ENTRIES) return;

  float cutf[11];
#pragma unroll
  for (int j = 0; j < 11; ++j) cutf[j] = (float)cutp[j];

  unsigned cs_pk[NUM_TABLES];
#pragma unroll
  for (int j = 0; j < NUM_TABLES; ++j) {
    unsigned short cb = __builtin_bit_cast(unsigned short, cutp[j]);
    half_t sh = (half_t)scalep[j];
    unsigned short sb = __builtin_bit_cast(unsigned short, sh);
    cs_pk[j] = (unsigned)cb | ((unsigned)sb << 16);
  }
  half_t t0h = (half_t)tablep[0];
  half_t tlh = (half_t)tablep[TABLE_N - 1];

  half_t v = __builtin_bit_cast(half_t, (unsigned short)b);
  float xf = (float)v;  // exact f32 of this fp16 pattern (inf/nan handled inside)
  float y = gelu_piecewise(xf, cutf, tablep, cs_pk, t0h, tlh);
  tab[b] = __builtin_bit_cast(unsigned short, (half_t)y);
}

// ---------------------------------------------------------------------------
// Main streaming kernel: clamp -> fp16 bits -> one ds_load_u16 gather.
// 128 KB table staged into STATIC LDS once per block via TDM
// (tensor_load_to_lds) or async global->LDS DMA fallback.
// ---------------------------------------------------------------------------
__device__ __forceinline__ float tab_one(float xf, float c0f, float clf,
                                         const unsigned short* __restrict__ s_tab)
{
  float xcl = __builtin_fminf(__builtin_fmaxf(xf, c0f), clf);
  half_t xc = (half_t)xcl;
  unsigned bits = (unsigned)__builtin_bit_cast(unsigned short, xc);
  bits = (xf != xf) ? 0u : bits;                // NaN -> gelu(0) entry
  return (float)__builtin_bit_cast(half_t, s_tab[bits]);  // ds_load_u16
}

__global__ __launch_bounds__(256) void gelu_tab_kernel(
    const float* __restrict__ x,
    const half_t* __restrict__ cutp,
    const unsigned short* __restrict__ tabg,
    float* __restrict__ out,
    long long nvec, long long n)
{
  // Static 128 KB LDS: legal only on CDNA5 (320 KB/WGP); compile-time checked.
  __shared__ unsigned short s_tab[TAB_ENTRIES];

#if __has_builtin(__builtin_amdgcn_tensor_load_to_lds)
  // --- Tensor Data Mover: one descriptor copies the whole 128 KB table ---
  if (threadIdx.x == 0) {
    unsigned lds_base = (unsigned)(size_t)&s_tab[0];          // LDS byte addr
    unsigned long long ga = (unsigned long long)(size_t)tabg; // global addr
    u32x4 g0;
    g0[0] = 1u;                                  // count=1, user descriptor
    g0[1] = lds_base;                            // lds_addr
    g0[2] = (unsigned)ga;                        // global_addr[31:0]
    g0[3] = (unsigned)(ga >> 32) | (2u << 30);   // global_addr[56:32] | type=2
    i32x8 g1;
    g1[0] = 0x00020000;                          // data_size=4B, mask=0
    g1[1] = (int)0x80000000u;                    // tensor_dim0 = 32768 dwords
    g1[2] = 0x00010000;                          // tensor_dim1 = 1
    g1[3] = (int)0x80000000u;                    // tile_dim0 = 32768 dwords
    g1[4] = 0;                                   // tile_dim1/2 unused
    g1[5] = 32768;                               // tensor_dim0_stride
    g1[6] = 0;
    g1[7] = 0;
    i32x4 gz; gz[0] = 0; gz[1] = 0; gz[2] = 0; gz[3] = 0;
    i32x8 gz8;
    gz8[0] = 0; gz8[1] = 0; gz8[2] = 0; gz8[3] = 0;
    gz8[4] = 0; gz8[5] = 0; gz8[6] = 0; gz8[7] = 0;
    // 6-arg toolchain variant: (g0, g1, g2, g3, extra_group, cpol)
    __builtin_amdgcn_tensor_load_to_lds(g0, g1, gz, gz, gz8, 0);
    __builtin_amdgcn_s_wait_tensorcnt(0);
  }
#else
  // --- Fallback: async global->LDS b128 DMA (ASYNCcnt path) ---
  {
    unsigned lds_base = (unsigned)(size_t)&s_tab[0];
    for (int off = (int)threadIdx.x * 16; off < TAB_BYTES; off += 256 * 16) {
      unsigned la = lds_base + (unsigned)off;
      unsigned go = (unsigned)off;
      asm volatile("global_load_async_to_lds_b128 %0, %1, %2"
                   :: "v"(la), "v"(go), "s"(tabg) : "memory");
    }
    asm volatile("s_wait_asynccnt 0" ::: "memory");
  }
#endif
  __syncthreads();

  const float c0f = (float)cutp[0];
  const float clf = (float)cutp[10];

  const long long stride = (long long)blockDim.x * gridDim.x;
  const f32x4* __restrict__ xv = (const f32x4*)x;
  f32x4* __restrict__ ov = (f32x4*)out;

  for (long long i = (long long)blockIdx.x * blockDim.x + threadIdx.x;
       i < nvec; i += 8 * stride) {
    f32x4 v[8];
    // 8 NT b128 loads in flight before any compute (latency cover at
    // 2-blocks/WGP occupancy imposed by the 128 KB LDS table)
#pragma unroll
    for (int k = 0; k < 8; ++k) {
      f32x4 t = 0.0f;
      long long ik = i + (long long)k * stride;
      if (ik < nvec) t = __builtin_nontemporal_load(&xv[ik]);
      v[k] = t;
    }
#pragma unroll
    for (int k = 0; k < 8; ++k) {
#pragma unroll
      for (int c = 0; c < 4; ++c) v[k][c] = tab_one(v[k][c], c0f, clf, s_tab);
    }
#pragma unroll
    for (int k = 0; k < 8; ++k) {
      long long ik = i + (long long)k * stride;
      if (ik < nvec) __builtin_nontemporal_store(v[k], &ov[ik]);
    }
  }

  // scalar tail (n % 4 elements)
  long long tail = nvec * 4;
  long long g = (long long)blockIdx.x * blockDim.x + threadIdx.x;
  if (g < n - tail) out[tail + g] = tab_one(x[tail + g], c0f, clf, s_tab);
}

// ---------------------------------------------------------------------------
// Fallback full-compute kernel (used only if ws_size < 128 KB): round-1 path.
// ---------------------------------------------------------------------------
__global__ __launch_bounds__(256) void gelu_lut_kernel(
    const float* __restrict__ x,
    const half_t* __restrict__ cutp,
    const float* __restrict__ tablep,
    const float* __restrict__ scalep,
    float* __restrict__ out,
    long long nvec, long long n)
{
  __shared__ float s_table[TABLE_N + 1];
  __shared__ unsigned s_cs[NUM_TABLES];

  const int tid = threadIdx.x;
  if (tid < TABLE_N) {
    unsigned lds_addr = (unsigned)(size_t)&s_table[tid];
    unsigned goff = (unsigned)(tid * 4);
    asm volatile("global_load_async_to_lds_b32 %0, %1, %2"
                 :: "v"(lds_addr), "v"(goff), "s"(tablep) : "memory");
  }
  if (tid < NUM_TABLES) {
    unsigned short cb = __builtin_bit_cast(unsigned short, cutp[tid]);
    half_t sh = (half_t)scalep[tid];
    unsigned short sb = __builtin_bit_cast(unsigned short, sh);
    s_cs[tid] = (unsigned)cb | ((unsigned)sb << 16);
  }
  asm volatile("s_wait_asynccnt 0" ::: "memory");
  __syncthreads();

  float cutf[11];
#pragma unroll
  for (int j = 0; j < 11; ++j) cutf[j] = (float)cutp[j];
  half_t t0h = (half_t)s_table[0];
  half_t tlh = (half_t)s_table[TABLE_N - 1];

  const long long stride = (long long)blockDim.x * gridDim.x;
  const f32x4* __restrict__ xv = (const f32x4*)x;
  f32x4* __restrict__ ov = (f32x4*)out;

  for (long long i = (long long)blockIdx.x * blockDim.x + tid; i < nvec;
       i += 4 * stride) {
    f32x4 v[4];
#pragma unroll
    for (int k = 0; k < 4; ++k) {
      f32x4 t = 0.0f;
      long long ik = i + (long long)k * stride;
      if (ik < nvec) t = __builtin_nontemporal_load(&xv[ik]);
      v[k] = t;
    }
#pragma unroll
    for (int k = 0; k < 4; ++k) {
#pragma unroll
      for (int c = 0; c < 4; ++c)
        v[k][c] = gelu_piecewise(v[k][c], cutf, s_table, s_cs, t0h, tlh);
    }
#pragma unroll
    for (int k = 0; k < 4; ++k) {
      long long ik = i + (long long)k * stride;
      if (ik < nvec) __builtin_nontemporal_store(v[k], &ov[ik]);
    }
  }

  long long tail = nvec * 4;
  long long g = (long long)blockIdx.x * blockDim.x + tid;
  if (g < n - tail)
    out[tail + g] = gelu_piecewise(x[tail + g], cutf, s_table, s_cs, t0h, tlh);
}

// ---------------------------------------------------------------------------
extern "C" void kernel_launch(void* const* d_in, const int* in_sizes, int n_in,
                              void* d_out, int out_size, void* d_ws, size_t ws_size,
                              hipStream_t stream) {
  (void)n_in; (void)in_sizes;
  const float*  x      = (const float*)d_in[0];
  const half_t* cutp   = (const half_t*)d_in[1];
  const float*  tablep = (const float*)d_in[2];
  const float*  scalep = (const float*)d_in[3];
  float* out = (float*)d_out;

  long long n    = (long long)out_size;
  long long nvec = n >> 2;
  const int block = 256;

  if (ws_size >= (size_t)TAB_BYTES) {
    unsigned short* tab = (unsigned short*)d_ws;
    build_tab_kernel<<<TAB_ENTRIES / block, block, 0, stream>>>(
        cutp, tablep, scalep, tab);

    long long per_block = (long long)block * 8;  // vec4 per block-iteration
    long long grid = (nvec + per_block - 1) / per_block;
    if (grid < 1) grid = 1;
    if (grid > 2048) grid = 2048;                // table restaging stays L2-hot
    gelu_tab_kernel<<<(int)grid, block, 0, stream>>>(
        x, cutp, tab, out, nvec, n);
  } else {
    long long per_block = (long long)block * 4;
    long long grid = (nvec + per_block - 1) / per_block;
    if (grid < 1) grid = 1;
    if (grid > 1048576) grid = 1048576;
    gelu_lut_kernel<<<(int)grid, block, 0, stream>>>(
        x, cutp, tablep, scalep, out, nvec, n);
  }
}